// _MojoEPReplicateCombineWrapper_45303315038644
// MI455X (gfx1250) — compile-verified
//
#include <hip/hip_runtime.h>
#include <cstdint>

// MoE combine for MI455X (gfx1250): sorted-index scatter-add inverted into a
// per-token gather (no atomics), streamed through the CDNA5 async
// global->LDS copy path with double buffering.
//
// Roofline: 512 MB total traffic / 23.3 TB/s ~= 22 us; 0.13 GFLOP is noise.
// Pure bandwidth problem -> b128 everywhere, async copies, NT stores.

#define AS1 __attribute__((address_space(1)))
#define AS3 __attribute__((address_space(3)))

typedef __attribute__((ext_vector_type(4))) int   i4;   // native 128-bit vectors
typedef __attribute__((ext_vector_type(4))) float f4;

typedef AS1 i4* gptr128;   // matches builtin param: V4i addrspace(1)*
typedef AS3 i4* lptr128;   // matches builtin param: V4i addrspace(3)*

// ---- CDNA5 async copy: global -> LDS (ASYNCcnt-tracked) ---------------------
__device__ __forceinline__ void async_copy_b128(const void* gsrc, void* ldst) {
#if __has_builtin(__builtin_amdgcn_global_load_async_to_lds_b128)
  __builtin_amdgcn_global_load_async_to_lds_b128(
      (gptr128)(uintptr_t)gsrc,
      (lptr128)(unsigned)(uintptr_t)ldst,
      /*offset=*/0, /*cpol=*/0);
#else
  unsigned lds_off = (unsigned)(uintptr_t)ldst;
  asm volatile("global_load_async_to_lds_b128 %0, %1, off"
               :: "v"(lds_off), "v"(gsrc)
               : "memory");
#endif
}

template <int N>
__device__ __forceinline__ void wait_asynccnt() {
#if __has_builtin(__builtin_amdgcn_s_wait_asynccnt)
  __builtin_amdgcn_s_wait_asynccnt(N);
#else
  asm volatile("s_wait_asynccnt %0" :: "i"(N) : "memory");
#endif
}

__device__ __forceinline__ void wait_dscnt0() {
#if __has_builtin(__builtin_amdgcn_s_wait_dscnt)
  __builtin_amdgcn_s_wait_dscnt(0);
#else
  asm volatile("s_wait_dscnt 0x0" ::: "memory");
#endif
}

// One block per (token, 1024-float column slice). 256 threads * f4.
__global__ __launch_bounds__(256) void moe_combine_kernel(
    const float* __restrict__ out_buf,   // [num_tokens, d_model]
    const float* __restrict__ expert,    // [num_rows,  d_model]
    const float* __restrict__ gates,     // [num_rows]
    const int*   __restrict__ tok_idx,   // [num_rows], sorted ascending
    float*       __restrict__ out,       // [num_tokens, d_model]
    int num_rows, int d_model)
{
  const int t   = blockIdx.x;                                   // token id (uniform)
  const int col = (blockIdx.y * blockDim.x + threadIdx.x) * 4;  // float index
  if (col >= d_model) return;

  // Segment [lo, lo+n) of rows whose token index == t.
  // t is uniform across the block -> SALU binary search; the 64 KB index
  // array lives in L2 after the first few blocks.
  int lo = 0, hi = num_rows;                 // lower_bound(t)
  while (lo < hi) { int m = (lo + hi) >> 1; if (tok_idx[m] <  t) lo = m + 1; else hi = m; }
  int lo2 = lo, hi2 = num_rows;              // upper_bound(t)
  while (lo2 < hi2) { int m = (lo2 + hi2) >> 1; if (tok_idx[m] <= t) lo2 = m + 1; else hi2 = m; }
  const int n = lo2 - lo;

  __shared__ f4 stage[2][256];               // 8 KB double buffer (LDS: 320 KB/WGP)

  const size_t base = (size_t)t * d_model + col;
  f4 acc = __builtin_nontemporal_load(
      reinterpret_cast<const f4*>(out_buf + base));

  // Prime the pipeline: up to two rows in flight per wave.
  if (n > 0) async_copy_b128(expert + (size_t)lo       * d_model + col, &stage[0][threadIdx.x]);
  if (n > 1) async_copy_b128(expert + (size_t)(lo + 1) * d_model + col, &stage[1][threadIdx.x]);

  for (int i = 0; i < n; ++i) {
    if (i + 1 < n) wait_asynccnt<1>();       // stage[i&1] complete, next still in flight
    else           wait_asynccnt<0>();

    const float g = gates[lo + i];           // uniform -> s_load
    f4 v = stage[i & 1][threadIdx.x];        // ds_load_b128, own slot -> no barrier
    acc.x = fmaf(g, v.x, acc.x);
    acc.y = fmaf(g, v.y, acc.y);
    acc.z = fmaf(g, v.z, acc.z);
    acc.w = fmaf(g, v.w, acc.w);

    // DS reads and async LDS writes are NOT mutually ordered on CDNA5:
    // drain DScnt before re-issuing an async write into this buffer.
    wait_dscnt0();
    if (i + 2 < n)
      async_copy_b128(expert + (size_t)(lo + i + 2) * d_model + col,
                      &stage[i & 1][threadIdx.x]);
  }

  // Written exactly once, never re-read -> non-temporal b128 store.
  __builtin_nontemporal_store(acc, reinterpret_cast<f4*>(out + base));
}

extern "C" void kernel_launch(void* const* d_in, const int* in_sizes, int n_in,
                              void* d_out, int out_size, void* d_ws, size_t ws_size,
                              hipStream_t stream) {
  // setup_inputs() order: output_buffer, expert_outputs, sorted_gates, token_indices
  const float* out_buf = (const float*)d_in[0];
  const float* expert  = (const float*)d_in[1];
  const float* gates   = (const float*)d_in[2];
  const int*   tok_idx = (const int*)d_in[3];   // harness delivers integer inputs as int32
  float* out = (float*)d_out;

  const int num_rows   = in_sizes[2];               // 16384
  const int d_model    = in_sizes[1] / num_rows;    // 4096
  const int num_tokens = out_size / d_model;        // 8192

  const int cols_per_block = 256 * 4;               // 1024 floats per block
  dim3 grid(num_tokens, (d_model + cols_per_block - 1) / cols_per_block);
  moe_combine_kernel<<<grid, 256, 0, stream>>>(out_buf, expert, gates, tok_idx,
                                               out, num_rows, d_model);
}